// RWSEEncoder_63256278335504
// MI455X (gfx1250) — compile-verified
//
#include <hip/hip_runtime.h>
#include <hip/hip_bf16.h>

#define PROBES   32
#define WALK     16
#define ODIM     8

typedef __attribute__((ext_vector_type(2))) float v2f;
typedef __attribute__((ext_vector_type(8))) float v8f;

// ---------------- utility kernels ----------------

__global__ void zero_f32_kernel(float* __restrict__ p, size_t n) {
    size_t i = (size_t)blockIdx.x * blockDim.x + threadIdx.x;
    if (i < n) p[i] = 0.0f;
}

__global__ void zero_u32_kernel(unsigned* __restrict__ p, size_t n) {
    size_t i = (size_t)blockIdx.x * blockDim.x + threadIdx.x;
    if (i < n) p[i] = 0u;
}

// ---------------- degree ----------------

__global__ void degree_kernel(const int* __restrict__ row, unsigned* __restrict__ deg, int E) {
    int e = blockIdx.x * blockDim.x + threadIdx.x;
    if (e < E) atomicAdd(&deg[row[e]], 1u);
}

__global__ void invdeg_kernel(const unsigned* __restrict__ deg, float* __restrict__ inv, int N) {
    int i = blockIdx.x * blockDim.x + threadIdx.x;
    if (i < N) {
        unsigned d = deg[i];
        inv[i] = 1.0f / (float)(d > 1u ? d : 1u);
    }
}

// ---------------- one sparse transition step ----------------
// grid.x covers edges (coalesced row/col loads), grid.y = probe chunk (8 chunks of 4)
__global__ void scatter_kernel(const int* __restrict__ row, const int* __restrict__ col,
                               const float* __restrict__ invdeg,
                               const float* __restrict__ p, float* __restrict__ pnew, int E) {
    int e = blockIdx.x * blockDim.x + threadIdx.x;
    if (e >= E) return;
    int c = (int)blockIdx.y << 2;
    int r  = row[e];
    int cl = col[e];
    float w = invdeg[r];
    const float4 v = *reinterpret_cast<const float4*>(p + (size_t)r * PROBES + c);
    float* dst = pnew + (size_t)cl * PROBES + c;
    atomicAdd(dst + 0, v.x * w);
    atomicAdd(dst + 1, v.y * w);
    atomicAdd(dst + 2, v.z * w);
    atomicAdd(dst + 3, v.w * w);
}

// ---------------- Hutchinson diagonal estimate for step k ----------------
__global__ void diag_kernel(const float* __restrict__ probes, const float* __restrict__ pnew,
                            float* __restrict__ rwdiag, int N, int k) {
    int n = blockIdx.x * blockDim.x + threadIdx.x;
    if (n >= N) return;
    const float4* a = reinterpret_cast<const float4*>(probes + (size_t)n * PROBES);
    const float4* b = reinterpret_cast<const float4*>(pnew   + (size_t)n * PROBES);
    float s = 0.0f;
#pragma unroll
    for (int j = 0; j < PROBES / 4; ++j) {
        float4 x = a[j];
        float4 y = b[j];
        s += x.x * y.x + x.y * y.y + x.z * y.z + x.w * y.w;
    }
    rwdiag[(size_t)n * WALK + k] = s * (1.0f / (float)PROBES);
}

// ---------------- final projection via V_WMMA_F32_16X16X4_F32 ----------------
// One wave handles a 16-node tile: D[16x16] = rw_diag_tile[16x16] * W^T (padded to 16 cols) + bias
// accumulated as 4 WMMAs with K=4.
__global__ void proj_wmma_kernel(const float* __restrict__ rwdiag, const float* __restrict__ W,
                                 const float* __restrict__ bias, float* __restrict__ out, int N) {
    int gtid = blockIdx.x * blockDim.x + threadIdx.x;
    int wave = gtid >> 5;                 // wave32
    int lane = threadIdx.x & 31;
    int ntiles = (N + 15) >> 4;
    if (wave >= ntiles) return;           // uniform per-wave exit -> EXEC all 1s below

    int m = lane & 15;                    // row within tile (A) / column index (B, C/D)
    int g = lane >> 4;                    // lane-group: selects K pair (A/B), row half (C/D)

    // C init: bias replicated down rows; columns >= ODIM padded with 0
    float bv = (m < ODIM) ? bias[m] : 0.0f;
    v8f c;
#pragma unroll
    for (int v = 0; v < 8; ++v) c[v] = bv;

    int node_m = wave * 16 + m;
    int rowm = node_m < N ? node_m : (N - 1);   // clamp reads; stores masked below

#pragma unroll
    for (int it = 0; it < 4; ++it) {
        int k0 = it * 4 + g * 2;          // A/B 32-bit layout: K = group*2 + vgpr
        v2f a, bb;
        a.x = rwdiag[(size_t)rowm * WALK + k0];
        a.y = rwdiag[(size_t)rowm * WALK + k0 + 1];
        // B[4x16]: column n = lane&15, B[k][n] = W[n][k], zero-padded for n >= ODIM
        bb.x = (m < ODIM) ? W[m * WALK + k0]     : 0.0f;
        bb.y = (m < ODIM) ? W[m * WALK + k0 + 1] : 0.0f;
        c = __builtin_amdgcn_wmma_f32_16x16x4_f32(
                /*neg_a=*/false, a, /*neg_b=*/false, bb,
                /*c_mod=*/(short)0, c, /*reuse_a=*/false, /*reuse_b=*/false);
    }

    // Store: C/D layout VGPR v -> row M = g*8 + v, column = lane&15 (only first ODIM cols real)
    if (m < ODIM) {
#pragma unroll
        for (int v = 0; v < 8; ++v) {
            int node = wave * 16 + g * 8 + v;
            if (node < N) out[(size_t)node * ODIM + m] = c[v];
        }
    }
}

// ---------------- host-side orchestration ----------------

extern "C" void kernel_launch(void* const* d_in, const int* in_sizes, int n_in,
                              void* d_out, int out_size, void* d_ws, size_t ws_size,
                              hipStream_t stream) {
    const int*   edge   = (const int*)d_in[0];     // [2, E] int32
    const float* probes = (const float*)d_in[2];   // [N, 32] f32
    const float* W      = (const float*)d_in[3];   // [8, 16] f32
    const float* bvec   = (const float*)d_in[4];   // [8]     f32

    const int E = in_sizes[0] / 2;
    const int N = in_sizes[2] / PROBES;
    const int* row = edge;
    const int* col = edge + E;

    // workspace layout (256B aligned)
    char* ws = (char*)d_ws;
    auto aln = [](size_t x) { return (x + 255) & ~(size_t)255; };
    size_t off = 0;
    unsigned* deg    = (unsigned*)(ws + off); off += aln((size_t)N * sizeof(unsigned));
    float*    invdeg = (float*)(ws + off);    off += aln((size_t)N * sizeof(float));
    float*    buf0   = (float*)(ws + off);    off += aln((size_t)N * PROBES * sizeof(float));
    float*    buf1   = (float*)(ws + off);    off += aln((size_t)N * PROBES * sizeof(float));
    float*    rwdiag = (float*)(ws + off);    off += aln((size_t)N * WALK * sizeof(float));
    (void)ws_size; (void)n_in; (void)out_size;

    const int B = 256;  // 8 wave32 waves per block

    // degree -> inverse degree
    zero_u32_kernel<<<(N + B - 1) / B, B, 0, stream>>>(deg, (size_t)N);
    degree_kernel<<<(E + B - 1) / B, B, 0, stream>>>(row, deg, E);
    invdeg_kernel<<<(N + B - 1) / B, B, 0, stream>>>(deg, invdeg, N);

    // 16 transition steps with L2-resident ping-pong state
    const float* cur = probes;
    float* nxt = buf0;
    const size_t cnt = (size_t)N * PROBES;
    for (int k = 0; k < WALK; ++k) {
        zero_f32_kernel<<<(unsigned)((cnt + B - 1) / B), B, 0, stream>>>(nxt, cnt);
        dim3 grid((E + B - 1) / B, PROBES / 4);
        scatter_kernel<<<grid, B, 0, stream>>>(row, col, invdeg, cur, nxt, E);
        diag_kernel<<<(N + B - 1) / B, B, 0, stream>>>(probes, nxt, rwdiag, N, k);
        cur = nxt;
        nxt = (cur == buf0) ? buf1 : buf0;
    }

    // final projection: one wave per 16-node tile, f32 WMMA
    int ntiles  = (N + 15) / 16;
    int threads = ntiles * 32;
    proj_wmma_kernel<<<(threads + B - 1) / B, B, 0, stream>>>(rwdiag, W, bvec, (float*)d_out, N);
}